// GlobalAttentionPool_89859305767779
// MI455X (gfx1250) — compile-verified
//
#include <hip/hip_runtime.h>
#include <hip/hip_bf16.h>

typedef __attribute__((ext_vector_type(2))) float v2f;
typedef __attribute__((ext_vector_type(8))) float v8f;

#define HIDDEN 256

// ---------------------------------------------------------------------------
// Kernel A: gate[i] = dot(h[i,:], w) + bias  via V_WMMA_F32_16X16X4_F32.
// One wave handles 16 rows; K swept 4 at a time (64 WMMAs / wave-tile).
// B operand = w replicated across all 16 columns -> every D column identical,
// so extraction only needs the documented C/D row layout.
// ---------------------------------------------------------------------------
__global__ __launch_bounds__(256) void gate_wmma_kernel(
    const float* __restrict__ h, const float* __restrict__ w,
    const float* __restrict__ bias_vec, float* __restrict__ gate, int N) {
  __shared__ float ws[HIDDEN];
  const int tid = threadIdx.x;
  ws[tid] = w[tid];           // blockDim.x == 256 == HIDDEN
  __syncthreads();

  const int wave = tid >> 5;
  const int lane = tid & 31;
  const long long tile = (long long)blockIdx.x * 8 + wave;  // 8 waves/block
  const long long row_base = tile * 16;
  if (row_base >= N) return;  // wave-uniform: EXEC stays all-ones for WMMA

  const int m = lane & 15;            // row within 16-row tile
  const int khalf = (lane >> 4) & 1;  // lanes 16..31 hold K+2,K+3
  long long row = row_base + m;
  if (row >= N) row = N - 1;          // clamp loads; store is guarded below

  const float* hp = h + row * (long long)HIDDEN + 2 * khalf;
  const float* wp = ws + 2 * khalf;

  v8f c = {};
#pragma unroll
  for (int kb = 0; kb < HIDDEN; kb += 4) {
    v2f a = *reinterpret_cast<const v2f*>(hp + kb);  // A[m][kb+2h : kb+2h+1]
    v2f b;
    b[0] = wp[kb];       // B[kb+2h][*]   (replicated over N columns)
    b[1] = wp[kb + 1];   // B[kb+2h+1][*]
    c = __builtin_amdgcn_wmma_f32_16x16x4_f32(
        /*neg_a=*/false, a, /*neg_b=*/false, b,
        /*c_mod=*/(short)0, c, /*reuse_a=*/false, /*reuse_b=*/false);
  }

  // C/D layout: VGPR v @ lane n(<16) = D[v][n]; @ lane 16+n = D[v+8][n].
  // All columns identical -> lane n<8 yields row n via c[n]; lane 16+n -> row 8+n.
  const float bias = bias_vec[0];
  const int r = lane & 7;
  float val = c[0];
#pragma unroll
  for (int k = 1; k < 8; ++k)
    if (r == k) val = c[k];
  if ((lane & 15) < 8) {
    long long orow = row_base + ((lane >> 4) << 3) + r;
    if (orow < N) gate[orow] = val + bias;
  }
}

// ---------------------------------------------------------------------------
// Kernel B: segment boundaries from the sorted batch vector.
// starts[s] = first i with batch[i] >= s ; starts[B] = N. Covers empty segs.
// ---------------------------------------------------------------------------
__global__ __launch_bounds__(256) void seg_starts_kernel(
    const int* __restrict__ batch, int* __restrict__ starts, int N, int B) {
  int i = blockIdx.x * blockDim.x + threadIdx.x;
  if (i >= N) return;
  int b = batch[i];
  int prev = (i == 0) ? -1 : batch[i - 1];
  for (int s = prev + 1; s <= b; ++s) starts[s] = i;
  if (i == N - 1)
    for (int s = b + 1; s <= B; ++s) starts[s] = N;
}

// ---------------------------------------------------------------------------
// Kernel C: per-segment softmax + weighted row sum. One block per segment,
// thread == channel (256). exp-weights staged in LDS per chunk; h rows are
// read fully coalesced (1 KB per row); prefetch hides the row stream.
// ---------------------------------------------------------------------------
__global__ __launch_bounds__(256) void seg_pool_kernel(
    const float* __restrict__ h, const float* __restrict__ gate,
    const int* __restrict__ starts, float* __restrict__ out) {
  constexpr int CHUNK = 1024;
  __shared__ float se[CHUNK];
  __shared__ float red[256];

  const int b = blockIdx.x;
  const int tid = threadIdx.x;
  const int s = starts[b];
  const int e = starts[b + 1];

  // ---- phase 1: segment max of gate ----
  float m = -INFINITY;
  for (int i = s + tid; i < e; i += 256) m = fmaxf(m, gate[i]);
  red[tid] = m;
  __syncthreads();
#pragma unroll
  for (int off = 128; off > 0; off >>= 1) {
    if (tid < off) red[tid] = fmaxf(red[tid], red[tid + off]);
    __syncthreads();
  }
  const float gmax = red[0];
  __syncthreads();

  // ---- phase 2+3: chunked exp -> LDS, then weighted row accumulation ----
  float acc = 0.f;
  float psum = 0.f;
  for (int cs = s; cs < e; cs += CHUNK) {
    const int ce = (cs + CHUNK < e) ? (cs + CHUNK) : e;
    for (int i = cs + tid; i < ce; i += 256) {
      float ev = __expf(gate[i] - gmax);
      se[i - cs] = ev;
      psum += ev;
    }
    __syncthreads();
    const int n = ce - cs;
    const float* hp = h + (long long)cs * HIDDEN + tid;
    for (int j = 0; j < n; ++j) {
      if (j + 16 < n)
        __builtin_prefetch(hp + (long long)(j + 16) * HIDDEN, 0, 1);
      acc += se[j] * hp[(long long)j * HIDDEN];
    }
    __syncthreads();
  }

  // ---- reduce exp-sum, normalize, write ----
  red[tid] = psum;
  __syncthreads();
#pragma unroll
  for (int off = 128; off > 0; off >>= 1) {
    if (tid < off) red[tid] += red[tid + off];
    __syncthreads();
  }
  const float inv = 1.f / (red[0] + 1e-8f);
  out[(long long)b * HIDDEN + tid] = acc * inv;
}

// ---------------------------------------------------------------------------
extern "C" void kernel_launch(void* const* d_in, const int* in_sizes, int n_in,
                              void* d_out, int out_size, void* d_ws,
                              size_t ws_size, hipStream_t stream) {
  const float* h      = (const float*)d_in[0];
  const float* gate_w = (const float*)d_in[1];
  const float* gate_b = (const float*)d_in[2];
  const int*   batch  = (const int*)d_in[3];

  const int N = in_sizes[0] / HIDDEN;   // h is (N, 256)
  const int B = out_size / HIDDEN;      // out is (B, 256)

  float* gate  = (float*)d_ws;                              // N floats
  int*   starts = (int*)((char*)d_ws + (size_t)N * sizeof(float));  // B+1 ints
  float* out   = (float*)d_out;

  // A: gate logits via WMMA. 16 rows/wave, 8 waves/block.
  const int tiles = (N + 15) / 16;
  const int blocksA = (tiles + 7) / 8;
  gate_wmma_kernel<<<blocksA, 256, 0, stream>>>(h, gate_w, gate_b, gate, N);

  // B: segment boundaries (batch is sorted).
  seg_starts_kernel<<<(N + 255) / 256, 256, 0, stream>>>(batch, starts, N, B);

  // C: per-segment softmax-weighted pooling.
  seg_pool_kernel<<<B, 256, 0, stream>>>(h, gate, starts, out);
}